// SGC_learnable_emb_Module_38242388803875
// MI455X (gfx1250) — compile-verified
//
#include <hip/hip_runtime.h>
#include <hip/hip_bf16.h>

#define N0 100000
#define N1 50000
#define N2 10000
#define E1C 1600000
#define E2C 320000
#define DD 128

typedef __attribute__((ext_vector_type(2))) float v2f;
typedef __attribute__((ext_vector_type(8))) float v8f;

// ---------------------------------------------------------------------------
// Zero the (N1+N2)*D workspace accumulators (graph-capture-safe memset).
// ---------------------------------------------------------------------------
__global__ void zero_ws_kernel(float4* __restrict__ p, int n4) {
    int i = blockIdx.x * blockDim.x + threadIdx.x;
    if (i < n4) p[i] = make_float4(0.f, 0.f, 0.f, 0.f);
}

// ---------------------------------------------------------------------------
// SpMM scatter: one wave32 per edge. Each lane loads a float4 of the source
// row (32 lanes * 16B = 512B = one full D=128 row), scales by the edge
// weight, and atomically accumulates into the destination row. x/h1/h2 are
// L2-resident on MI455X (192MB L2), so the atomics stay on-chip.
// ---------------------------------------------------------------------------
__global__ void spmm_scatter_kernel(const float* __restrict__ h_src,
                                    const int* __restrict__ src,
                                    const int* __restrict__ dst,
                                    const float* __restrict__ ew,
                                    float* __restrict__ h_dst,
                                    int num_edges) {
    int e    = (blockIdx.x * (blockDim.x >> 5)) + (threadIdx.x >> 5);
    int lane = threadIdx.x & 31;
    if (e >= num_edges) return;

    int   s = __builtin_amdgcn_readfirstlane(src[e]);
    int   d = __builtin_amdgcn_readfirstlane(dst[e]);
    float w = ew[e];

    const float4* row = (const float4*)(h_src + (size_t)s * DD);
    float4 v = row[lane];

    float* out = h_dst + (size_t)d * DD + lane * 4;
    __hip_atomic_fetch_add(out + 0, v.x * w, __ATOMIC_RELAXED, __HIP_MEMORY_SCOPE_AGENT);
    __hip_atomic_fetch_add(out + 1, v.y * w, __ATOMIC_RELAXED, __HIP_MEMORY_SCOPE_AGENT);
    __hip_atomic_fetch_add(out + 2, v.z * w, __ATOMIC_RELAXED, __HIP_MEMORY_SCOPE_AGENT);
    __hip_atomic_fetch_add(out + 3, v.w * w, __ATOMIC_RELAXED, __HIP_MEMORY_SCOPE_AGENT);
}

// ---------------------------------------------------------------------------
// out = h2 @ W^T + b using V_WMMA_F32_16X16X4_F32 (full fp32 precision).
// One wave per 16x16 output tile; K swept in steps of 4 (32 WMMAs/tile).
//
// Fragment layouts per CDNA5 ISA 7.12.2:
//   A (16x4 f32):  lanes 0-15 -> M=lane, VGPR{0,1}=K{k,k+1};
//                  lanes 16-31 -> M=lane-16, VGPR{0,1}=K{k+2,k+3}
//   B (4x16 f32):  B[kk][n] = W[n0+n][k+kk]  (since B = W^T slice), which
//                  gives the mirrored per-lane addressing: lane r holds
//                  column n=r, halves select K pairs {k,k+1}/{k+2,k+3}.
//   C/D (16x16 f32, 8 VGPRs): VGPR i -> M=i (lanes 0-15), M=i+8 (lanes 16-31),
//                  N = lane & 15.
// ---------------------------------------------------------------------------
__global__ void gemm_bias_wmma_kernel(const float* __restrict__ A,    // h2 [N2, D]
                                      const float* __restrict__ Wm,   // W  [D, D]
                                      const float* __restrict__ bias, // b  [D]
                                      float* __restrict__ out) {      // [N2, D]
    const int tiles = (N2 / 16) * (DD / 16);  // 625 * 8 = 5000 waves
    int wid = (blockIdx.x * (blockDim.x >> 5)) + (threadIdx.x >> 5);
    if (wid >= tiles) return;

    int mt = wid >> 3;        // 8 n-tiles per m-tile row
    int nt = wid & 7;
    int lane = threadIdx.x & 31;
    int half = lane >> 4;     // 0 or 1
    int r    = lane & 15;
    int m0 = mt * 16, n0 = nt * 16;

    const float* arow = A  + (size_t)(m0 + r) * DD + half * 2;
    const float* brow = Wm + (size_t)(n0 + r) * DD + half * 2;
    float bval = bias[n0 + r];

    v8f c = {};
#pragma unroll
    for (int k = 0; k < DD; k += 4) {
        v2f a, b;
        a.x = arow[k];
        a.y = arow[k + 1];
        b.x = brow[k];
        b.y = brow[k + 1];
        // 8 args: (neg_a, A, neg_b, B, c_mod, C, reuse_a, reuse_b)
        c = __builtin_amdgcn_wmma_f32_16x16x4_f32(
                false, a, false, b, (short)0, c, false, false);
    }

    float* orow = out + (size_t)(m0 + half * 8) * DD + n0 + r;
#pragma unroll
    for (int i = 0; i < 8; ++i) {
        orow[(size_t)i * DD] = c[i] + bval;
    }
}

// ---------------------------------------------------------------------------
// Host launcher. Input order (setup_inputs dict order):
//   0:x 1:ew1 2:ew2 3:W 4:b 5:e1_src 6:e1_dst 7:e2_src 8:e2_dst
// d_out: [N2, D] fp32. Workspace: h1 (N1*D floats) | h2 (N2*D floats).
// ---------------------------------------------------------------------------
extern "C" void kernel_launch(void* const* d_in, const int* in_sizes, int n_in,
                              void* d_out, int out_size, void* d_ws, size_t ws_size,
                              hipStream_t stream) {
    const float* x      = (const float*)d_in[0];
    const float* ew1    = (const float*)d_in[1];
    const float* ew2    = (const float*)d_in[2];
    const float* W      = (const float*)d_in[3];
    const float* b      = (const float*)d_in[4];
    const int*   e1_src = (const int*)d_in[5];
    const int*   e1_dst = (const int*)d_in[6];
    const int*   e2_src = (const int*)d_in[7];
    const int*   e2_dst = (const int*)d_in[8];
    float*       out    = (float*)d_out;

    float* h1 = (float*)d_ws;                 // [N1, D]
    float* h2 = h1 + (size_t)N1 * DD;         // [N2, D]

    // 1) zero accumulators
    {
        int n4 = (N1 + N2) * DD / 4;
        int blocks = (n4 + 255) / 256;
        zero_ws_kernel<<<blocks, 256, 0, stream>>>((float4*)d_ws, n4);
    }
    // 2) SpMM block 1: x -> h1 (8 edges per 256-thread block)
    {
        int blocks = (E1C + 7) / 8;
        spmm_scatter_kernel<<<blocks, 256, 0, stream>>>(x, e1_src, e1_dst, ew1, h1, E1C);
    }
    // 3) SpMM block 2: h1 -> h2
    {
        int blocks = (E2C + 7) / 8;
        spmm_scatter_kernel<<<blocks, 256, 0, stream>>>(h1, e2_src, e2_dst, ew2, h2, E2C);
    }
    // 4) out = h2 @ W^T + b via fp32 WMMA (5000 tile-waves, 8 waves/block)
    {
        int tiles  = (N2 / 16) * (DD / 16);
        int blocks = (tiles + 7) / 8;
        gemm_bias_wmma_kernel<<<blocks, 256, 0, stream>>>(h2, W, b, out);
    }
}